// SpikingActivation_44392781972088
// MI455X (gfx1250) — compile-verified
//
#include <hip/hip_runtime.h>
#include <stdint.h>

#define B_      32
#define T_      2048
#define U_      512
#define NCHUNK  16
#define TC      128            // T_ / NCHUNK
#define STAGE   8              // rows staged into LDS per step
#define NSTAGES (TC / STAGE)   // 16
#define ISSUES  2              // async b128 issues per stage (block covers 4 rows/issue)
#define DT_F    0.001f
#define INV_DT  1000.0f

// ---------------------------------------------------------------------------
// Kernel 1: per-chunk sums of relu(x)*dt  ->  ws[b][chunk][u]
// One thread per (b, chunk, 4 consecutive u). 65536 threads, float4 loads.
// ---------------------------------------------------------------------------
__global__ __launch_bounds__(256)
void k1_chunk_sums(const float* __restrict__ x, float* __restrict__ ws) {
    int g  = blockIdx.x * 256 + threadIdx.x;   // 0 .. 65535
    int u4 = g & 127;                          // u = u4*4
    int c  = (g >> 7) & (NCHUNK - 1);
    int b  = g >> 11;

    const float4* p = (const float4*)(x + (size_t)(b * T_ + c * TC) * U_) + u4;
    float4 acc = {0.f, 0.f, 0.f, 0.f};
#pragma unroll 4
    for (int t = 0; t < TC; ++t) {
        float4 v = p[(size_t)t * (U_ / 4)];
        acc.x += fmaxf(v.x, 0.f) * DT_F;
        acc.y += fmaxf(v.y, 0.f) * DT_F;
        acc.z += fmaxf(v.z, 0.f) * DT_F;
        acc.w += fmaxf(v.w, 0.f) * DT_F;
    }
    ((float4*)(ws + (size_t)(b * NCHUNK + c) * U_))[u4] = acc;
}

// ---------------------------------------------------------------------------
// Kernel 2: exclusive prefix over chunks, seeded with v0.
// After this, ws[b][c][u] = C value at entry of chunk c for channel (b,u).
// ---------------------------------------------------------------------------
__global__ __launch_bounds__(256)
void k2_prefix(const float* __restrict__ v0, float* __restrict__ ws) {
    int g = blockIdx.x * 256 + threadIdx.x;    // 0 .. 16383
    int u = g & (U_ - 1);
    int b = g >> 9;
    float run = v0[(size_t)b * U_ + u];
#pragma unroll
    for (int c = 0; c < NCHUNK; ++c) {
        size_t idx = (size_t)(b * NCHUNK + c) * U_ + u;
        float t = ws[idx];
        ws[idx] = run;
        run += t;
    }
}

// ---------------------------------------------------------------------------
// Kernel 3: sequential scan within each chunk, staged through LDS with
// double-buffered async global->LDS copies (CDNA5 GLOBAL_LOAD_ASYNC_TO_LDS).
// Block = (b, chunk, half-of-U): 128 rows x 256 cols tile.
// ---------------------------------------------------------------------------
__global__ __launch_bounds__(256)
void k3_scan(const float* __restrict__ x, const float* __restrict__ ws,
             float* __restrict__ out) {
    __shared__ __align__(16) float buf[2][STAGE * 256];   // 2 x 8 KiB

    const int tid  = threadIdx.x;
    const int blk  = blockIdx.x;           // 1024 = b(32) * chunk(16) * half(2)
    const int half = blk & 1;
    const int c    = (blk >> 1) & (NCHUNK - 1);
    const int b    = blk >> 5;

    const size_t rowBase = (size_t)(b * T_ + c * TC) * U_ + half * 256;
    const float* baseG   = x + rowBase;                    // uniform -> SGPR pair

    // Per-lane async-copy mapping: each issue moves 4 rows (256 lanes x 16B).
    const unsigned rowInIssue = (unsigned)(tid >> 6);       // 0..3
    const unsigned colByte    = (unsigned)(tid & 63) * 16u; // 0..1008

    // Scan state: C at chunk entry; previous floor is floor of entry value
    // (== floor(C_{t-1}) at the chunk boundary; floor(v0) for chunk 0).
    const int u = half * 256 + tid;
    float C  = ws[(size_t)(b * NCHUNK + c) * U_ + u];
    float pf = floorf(C);

    float* outBase = out + rowBase + tid;

    auto prefetch = [&](int s) {
        unsigned ldsBase = (unsigned)(uintptr_t)(&buf[s & 1][0]);
#pragma unroll
        for (int i = 0; i < ISSUES; ++i) {
            unsigned row   = (unsigned)(s * STAGE + i * 4) + rowInIssue;
            unsigned goff  = row * (unsigned)(U_ * 4) + colByte;           // global byte off
            unsigned laddr = ldsBase + ((unsigned)(i * 4) + rowInIssue) * 1024u + colByte;
            asm volatile("global_load_async_to_lds_b128 %0, %1, %2"
                         :
                         : "v"(laddr), "v"(goff), "s"(baseG)
                         : "memory");
        }
    };

    prefetch(0);
    for (int s = 0; s < NSTAGES; ++s) {
        if (s + 1 < NSTAGES) {
            // Target buffer held stage s-1; its consumption finished at the
            // trailing barrier of the previous iteration -> safe to overwrite.
            prefetch(s + 1);
            // <=2 outstanding: the 2 issues of stage s+1 may still be in
            // flight; the (in-order) stage-s issues are guaranteed complete.
            asm volatile("s_wait_asynccnt 0x2" ::: "memory");
        } else {
            asm volatile("s_wait_asynccnt 0x0" ::: "memory");
        }
        __syncthreads();   // all waves' stage-s data now in LDS

        const float* src = &buf[s & 1][0];
#pragma unroll
        for (int r = 0; r < STAGE; ++r) {
            float v = src[r * 256 + tid];
            C += fmaxf(v, 0.f) * DT_F;
            float f = floorf(C);
            __builtin_nontemporal_store((f - pf) * INV_DT,
                                        outBase + (size_t)(s * STAGE + r) * U_);
            pf = f;
        }
        __syncthreads();   // stage-s buffer free before prefetch(s+2) reuses it
    }
}

// ---------------------------------------------------------------------------
extern "C" void kernel_launch(void* const* d_in, const int* in_sizes, int n_in,
                              void* d_out, int out_size, void* d_ws, size_t ws_size,
                              hipStream_t stream) {
    const float* x  = (const float*)d_in[0];   // [32, 2048, 512] f32
    const float* v0 = (const float*)d_in[1];   // [32, 512] f32
    float* out = (float*)d_out;                // [32, 2048, 512] f32
    float* ws  = (float*)d_ws;                 // needs 32*16*512*4 = 1 MiB

    k1_chunk_sums<<<dim3(256),  dim3(256), 0, stream>>>(x, ws);
    k2_prefix   <<<dim3(64),   dim3(256), 0, stream>>>(v0, ws);
    k3_scan     <<<dim3(1024), dim3(256), 0, stream>>>(x, ws, out);
}